// GCNNet_89094801588988
// MI455X (gfx1250) — compile-verified
//
#include <hip/hip_runtime.h>
#include <hip/hip_bf16.h>

#define N_NODES 50000
#define DFEAT   128
#define N_EDGES 640000

typedef __attribute__((ext_vector_type(2))) float v2f;
typedef __attribute__((ext_vector_type(8))) float v8f;

// ---------------- utility kernels ----------------

__global__ void zero_f32(float* __restrict__ p, int n) {
    int i = blockIdx.x * blockDim.x + threadIdx.x;
    if (i < n) p[i] = 0.0f;
}

__global__ void degree_kernel(const long long* __restrict__ dst,
                              float* __restrict__ deg, int nE) {
    int e = blockIdx.x * blockDim.x + threadIdx.x;
    if (e < nE) atomicAdd(deg + dst[e], 1.0f);
}

__global__ void norm_kernel(const float* __restrict__ deg,
                            float* __restrict__ nrm, int n) {
    int i = blockIdx.x * blockDim.x + threadIdx.x;
    if (i < n) nrm[i] = rsqrtf(fmaxf(deg[i], 1.0f));
}

// hs[n,d] = h[n,d] * nrm[n]   (pre-scale once, gather reuses it ~12.8x)
__global__ void prescale_kernel(const float* __restrict__ h,
                                const float* __restrict__ nrm,
                                float* __restrict__ hs, int n) {
    int i = blockIdx.x * blockDim.x + threadIdx.x;
    if (i < n) hs[i] = h[i] * nrm[i >> 7];   // DFEAT == 128
}

// one wave32 per edge: lane moves float4 (4*32 = 128 floats = full row)
__global__ void scatter_kernel(const float* __restrict__ hs,
                               const long long* __restrict__ src,
                               const long long* __restrict__ dst,
                               float* __restrict__ agg, int nE) {
    int gwave = (int)((blockIdx.x * blockDim.x + threadIdx.x) >> 5);
    int lane  = threadIdx.x & 31;
    if (gwave >= nE) return;
    long long s = src[gwave];
    long long d = dst[gwave];
    const float4 v = *(const float4*)(hs + s * DFEAT + lane * 4);
    float* o = agg + d * DFEAT + lane * 4;
    atomicAdd(o + 0, v.x);
    atomicAdd(o + 1, v.y);
    atomicAdd(o + 2, v.z);
    atomicAdd(o + 3, v.w);
}

// ---------------- fused post-scale + GEMM (+ReLU) via WMMA fp32 ----------------
// out[row, 0:NOUT] = act( (A[row,:]*nrm[row]) @ W[128, NOUT] )
// block = 256 threads = 8 waves; block handles 16 rows; wave w owns cols [16w,16w+16)
template <int NOUT, bool RELU>
__global__ void gemm_kernel(const float* __restrict__ A,
                            const float* __restrict__ nrm,
                            const float* __restrict__ W,
                            float* __restrict__ out) {
    __shared__ float lds[16 * 132];            // 16x128 A-tile, stride 132 (bank-safe)
    const int t    = threadIdx.x;
    const int row0 = blockIdx.x * 16;

    // stage norm-scaled A tile (post-scale fused here)
#pragma unroll
    for (int i = 0; i < 8; ++i) {
        int idx = t + i * 256;                 // 0 .. 2047
        int r = idx >> 7, c = idx & 127;
        lds[r * 132 + c] = A[(size_t)(row0 + r) * DFEAT + c] * nrm[row0 + r];
    }
    __syncthreads();

    const int wave = t >> 5;
    const int lane = t & 31;
    if (wave * 16 >= NOUT) return;             // uniform per-wave exit (EXEC stays full)

    const int half = lane >> 4;                // 0: lanes 0-15, 1: lanes 16-31
    const int l16  = lane & 15;
    const int col  = wave * 16 + l16;
    const int colc = col < NOUT ? col : NOUT - 1;
    const float cmask = (col < NOUT) ? 1.0f : 0.0f;   // branchless OOB handling

    v8f acc = {};
#pragma unroll 4
    for (int kb = 0; kb < DFEAT; kb += 4) {
        const int k = kb + 2 * half;           // A frag: K = j + 2*(lane/16), M = l16
        v2f a = *(const v2f*)(&lds[l16 * 132 + k]);
        v2f b;                                  // B frag: K = j + 2*(lane/16), N = l16
        b.x = W[(size_t)k       * NOUT + colc] * cmask;
        b.y = W[(size_t)(k + 1) * NOUT + colc] * cmask;
        acc = __builtin_amdgcn_wmma_f32_16x16x4_f32(
            false, a, false, b, (short)0, acc, false, false);
    }

    // C/D layout: VGPR r -> M = r + 8*(lane/16), N = l16
#pragma unroll
    for (int r = 0; r < 8; ++r) {
        int row = row0 + r + 8 * half;
        float v = acc[r];
        if (RELU) v = fmaxf(v, 0.0f);
        if (col < NOUT) out[(size_t)row * NOUT + col] = v;
    }
}

// ---------------- driver ----------------

extern "C" void kernel_launch(void* const* d_in, const int* in_sizes, int n_in,
                              void* d_out, int out_size, void* d_ws, size_t ws_size,
                              hipStream_t stream) {
    (void)in_sizes; (void)n_in; (void)out_size; (void)ws_size;

    const float*      features = (const float*)d_in[0];
    const long long*  ei       = (const long long*)d_in[1];   // int64 [2, E]
    const float*      Wl[4]    = {(const float*)d_in[2], (const float*)d_in[3],
                                  (const float*)d_in[4], (const float*)d_in[5]};
    float* out = (float*)d_out;

    const long long* src = ei;
    const long long* dst = ei + N_EDGES;

    // workspace layout
    float* deg  = (float*)d_ws;
    float* nrm  = deg + N_NODES;
    float* hs   = nrm + N_NODES;                       // 50000*128
    float* agg  = hs  + (size_t)N_NODES * DFEAT;       // 50000*128
    float* hout = agg + (size_t)N_NODES * DFEAT;       // 50000*128

    const int TB = 256;
    const int nElem = N_NODES * DFEAT;                 // 6.4M

    // degree + norm
    zero_f32<<<(N_NODES + TB - 1) / TB, TB, 0, stream>>>(deg, N_NODES);
    degree_kernel<<<(N_EDGES + TB - 1) / TB, TB, 0, stream>>>(dst, deg, N_EDGES);
    norm_kernel<<<(N_NODES + TB - 1) / TB, TB, 0, stream>>>(deg, nrm, N_NODES);

    const float* h = features;
    for (int l = 0; l < 4; ++l) {
        prescale_kernel<<<(nElem + TB - 1) / TB, TB, 0, stream>>>(h, nrm, hs, nElem);
        zero_f32<<<(nElem + TB - 1) / TB, TB, 0, stream>>>(agg, nElem);
        // one wave per edge -> N_EDGES*32 threads
        scatter_kernel<<<(N_EDGES * 32) / TB, TB, 0, stream>>>(hs, src, dst, agg, N_EDGES);

        if (l < 3) {
            gemm_kernel<128, true><<<N_NODES / 16, TB, 0, stream>>>(agg, nrm, Wl[l], hout);
            h = hout;
        } else {
            gemm_kernel<40, false><<<N_NODES / 16, TB, 0, stream>>>(agg, nrm, Wl[l], out);
        }
    }
}